// MultiTaskProbabilisticAutoma_19344532701882
// MI455X (gfx1250) — compile-verified
//
#include <hip/hip_runtime.h>

typedef float v2f __attribute__((ext_vector_type(2)));
typedef float v8f __attribute__((ext_vector_type(8)));

#define BB 256
#define LL 512
#define AA 16
#define SS 64
#define RR 2
#define CHUNK 16
#define NCHUNK (LL / CHUNK)   // 32 chunks of 16 timesteps

// One workgroup per batch element. 256 threads = 8 wave32.
// Phase 1 (per chunk): WMMA GEMM  M[t, s*64+n] = sum_a act[t0+t, a] * T[a, s*64+n]
//   -> M = 16 (timesteps), N = 4096 (s,n flattened), K = 16 (a), via 4x K=4 f32 WMMA.
// Phase 2 (per chunk): sequential scan s_{t+1} = s_t @ M_t, rewards r_t = s_{t+1} @ rew.
__global__ __launch_bounds__(256)
void automaton_wmma_kernel(const float* __restrict__ act,    // [B, L, A]
                           const float* __restrict__ trans,  // [B, A, S, S]
                           const float* __restrict__ rew,    // [B, S, R]
                           float* __restrict__ out_states,   // [B, L, S]
                           float* __restrict__ out_rew)      // [B, L, R]
{
    __shared__ float M_lds[CHUNK * SS * SS];   // 16 x 4096 f32 = 256 KB
    __shared__ float s_lds[SS];                // current state distribution
    __shared__ float red[4 * SS];              // 4-way partial reduction over s
    __shared__ float rred[4];                  // reward partials

    const int b    = blockIdx.x;
    const int tid  = threadIdx.x;
    const int lane = tid & 31;
    const int wave = tid >> 5;

    const float* Tb   = trans + (size_t)b * AA * SS * SS;  // T[a, c], c = s*64+n
    const float* actb = act   + (size_t)b * LL * AA;
    const float* rewb = rew   + (size_t)b * SS * RR;

    // initial state: one-hot on state 0
    if (tid < SS) s_lds[tid] = (tid == 0) ? 1.0f : 0.0f;
    __syncthreads();

    // WMMA 32-bit operand striping (wave32):
    //  A 16x4 : VGPR0 = K{0|2}, VGPR1 = K{1|3}; M = lane&15; hi-half lanes take K+2
    //  B 4x16 : VGPR0 = row K{0|2}, VGPR1 = row K{1|3}; N = lane&15
    //  D 16x16: VGPR i = row i (lanes 0-15) / row i+8 (lanes 16-31); N = lane&15
    const int colL   = lane & 15;
    const int rowoff = (lane < 16) ? 0 : 2;
    const int r0     = (lane < 16) ? 0 : 8;

    for (int chunk = 0; chunk < NCHUNK; ++chunk) {
        const int t0 = chunk * CHUNK;

        // ---- load A tile (act 16 timesteps x 16 actions) in WMMA layout ----
        v2f amat[4];
        {
            const int trow = lane & 15;
            #pragma unroll
            for (int ks = 0; ks < 4; ++ks) {
                amat[ks].x = actb[(size_t)(t0 + trow) * AA + 4 * ks + rowoff];
                amat[ks].y = actb[(size_t)(t0 + trow) * AA + 4 * ks + rowoff + 1];
            }
        }

        // ---- GEMM: wave covers N-columns [wave*512, wave*512+512), 32 tiles ----
        for (int jt = 0; jt < 32; ++jt) {
            const int c0 = wave * 512 + jt * 16;
            v8f acc = {0.f, 0.f, 0.f, 0.f, 0.f, 0.f, 0.f, 0.f};
            #pragma unroll
            for (int ks = 0; ks < 4; ++ks) {
                v2f bmat;
                bmat.x = Tb[(size_t)(4 * ks + rowoff)     * (SS * SS) + c0 + colL];
                bmat.y = Tb[(size_t)(4 * ks + rowoff + 1) * (SS * SS) + c0 + colL];
                acc = __builtin_amdgcn_wmma_f32_16x16x4_f32(
                        false, amat[ks], false, bmat, (short)0, acc, false, false);
            }
            // scatter D tile into LDS: M_lds[t][c0 + colL]
            const int nc = c0 + colL;
            #pragma unroll
            for (int i = 0; i < 8; ++i)
                M_lds[(r0 + i) * (SS * SS) + nc] = acc[i];
        }
        __syncthreads();

        // ---- sequential scan over the 16 timesteps of this chunk ----
        const int n    = tid & 63;   // output state index
        const int part = tid >> 6;   // 4-way partition of the s-contraction
        for (int tt = 0; tt < CHUNK; ++tt) {
            const float* Mrow = &M_lds[tt * (SS * SS) + n];
            float p = 0.0f;
            #pragma unroll
            for (int si = 0; si < 16; ++si) {
                const int s = part * 16 + si;
                p += s_lds[s] * Mrow[s * SS];
            }
            red[part * SS + n] = p;
            __syncthreads();

            float ns = 0.0f;
            if (tid < SS) {
                ns = red[n] + red[SS + n] + red[2 * SS + n] + red[3 * SS + n];
                out_states[((size_t)b * LL + t0 + tt) * SS + n] = ns;
            }
            __syncthreads();
            if (tid < SS) s_lds[n] = ns;
            __syncthreads();

            // rewards: r[j] = sum_n s_lds[n] * rew[n, j]; waves 0-3, shfl reduce
            if (tid < 2 * SS) {
                const int j  = tid >> 6;
                float pr = s_lds[n] * rewb[n * RR + j];
                #pragma unroll
                for (int off = 16; off >= 1; off >>= 1)
                    pr += __shfl_down(pr, off, 32);
                if (lane == 0) rred[wave] = pr;
            }
            __syncthreads();
            if (tid == 0) {
                const size_t ro = ((size_t)b * LL + t0 + tt) * RR;
                out_rew[ro + 0] = rred[0] + rred[1];
                out_rew[ro + 1] = rred[2] + rred[3];
            }
            __syncthreads();
        }
        // final barrier of the tt loop also guards M_lds reuse next chunk
    }
}

extern "C" void kernel_launch(void* const* d_in, const int* in_sizes, int n_in,
                              void* d_out, int out_size, void* d_ws, size_t ws_size,
                              hipStream_t stream) {
    (void)in_sizes; (void)n_in; (void)out_size; (void)d_ws; (void)ws_size;
    const float* act   = (const float*)d_in[0];  // [B, L, A]
    const float* trans = (const float*)d_in[1];  // [B, A, S, S]
    const float* rew   = (const float*)d_in[2];  // [B, S, R]
    float* out_states = (float*)d_out;                         // [B, L, S]
    float* out_rew    = out_states + (size_t)BB * LL * SS;     // [B, L, R]
    automaton_wmma_kernel<<<dim3(BB), dim3(256), 0, stream>>>(
        act, trans, rew, out_states, out_rew);
}